// ContextualAttention_74612171866194
// MI455X (gfx1250) — compile-verified
//
#include <hip/hip_runtime.h>
#include <hip/hip_bf16.h>
#include <math.h>

// ---------------- problem constants ----------------
#define B_   2
#define C_   256
#define H_   48
#define W_   48
#define HD   24
#define WD   24
#define L_   576          // HD*WD positions / filters
#define K9   2304         // C*9  (3x3 patch feature dim)
#define K16  4096         // C*16 (4x4 raw patch dim)
#define SCALE_ 10.0f
#define EPS_   1e-4f

typedef float v2f __attribute__((ext_vector_type(2)));
typedef float v8f __attribute__((ext_vector_type(8)));

#define TK  32            // K-chunk staged in LDS
#define LDP (TK + 2)      // padded LDS row (keeps 8B alignment, spreads banks)

// ---- gather: 3x3 same-padded stride-1 patches of a downsampled [C,24,24] tensor
__device__ __forceinline__ float patch9(const float* ds, int b, int k, int l) {
    int c = k / 9, r = k % 9;
    int ki = r / 3, kj = r % 3;
    int y = l / WD + ki - 1;
    int x = l % WD + kj - 1;
    if ((unsigned)y >= HD || (unsigned)x >= WD) return 0.0f;
    return ds[((b * C_ + c) * HD + y) * WD + x];
}

// ---- gather: 4x4 stride-2 pad-1 raw patches of a full-res [C,48,48] tensor
__device__ __forceinline__ float patch16(const float* sc, int b, int m, int l) {
    int c = m >> 4, i = (m >> 2) & 3, j = m & 3;
    int r = 2 * (l / WD) + i - 1;
    int s = 2 * (l % WD) + j - 1;
    if ((unsigned)r >= H_ || (unsigned)s >= W_) return 0.0f;
    return sc[((b * C_ + c) * H_ + r) * W_ + s];
}

// ---------------- kernel 1: nearest-neighbor downsample x0.5 ----------------
__global__ __launch_bounds__(256) void k_prep_ds(const float* __restrict__ left,
                                                 const float* __restrict__ right,
                                                 const float* __restrict__ mid,
                                                 float* __restrict__ dsL,
                                                 float* __restrict__ dsR,
                                                 float* __restrict__ dsM) {
    int idx = blockIdx.x * 256 + threadIdx.x;
    if (idx >= B_ * C_ * HD * WD) return;
    int x = idx % WD, t = idx / WD;
    int y = t % HD;  t /= HD;
    int c = t % C_;  int b = t / C_;
    int src = ((b * C_ + c) * H_ + 2 * y) * W_ + 2 * x;
    dsL[idx] = left[src];
    dsR[idx] = right[src];
    dsM[idx] = mid[src];
}

// ---------------- kernel 2: per-patch L2 denominators ----------------
// denom[b,l] = sqrt( sum_k fp^2 + K9*EPS )   (EPS added per element in ref)
__global__ __launch_bounds__(256) void k_denom(const float* __restrict__ dsL,
                                               const float* __restrict__ dsR,
                                               float* __restrict__ denL,
                                               float* __restrict__ denR) {
    const float* ds = blockIdx.y ? dsR : dsL;
    float* den      = blockIdx.y ? denR : denL;
    int gl   = blockIdx.x * 8 + (threadIdx.x >> 5);   // (b*L + l), 0..1151
    int b    = gl / L_;
    int l    = gl % L_;
    int lane = threadIdx.x & 31;
    float s = 0.0f;
    for (int k = lane; k < K9; k += 32) {
        float v = patch9(ds, b, k, l);
        s += v * v;
    }
    #pragma unroll
    for (int off = 16; off; off >>= 1) s += __shfl_xor(s, off, 32);
    if (lane == 0) den[gl] = sqrtf(s + (float)K9 * EPS_);
}

// ---------------- kernel 3: scores GEMM (fp32 WMMA 16x16x4) ----------------
// scores[b,l,p] = SCALE/denom[l] * sum_k patch(feat)[k,l] * patch(mid)[k,p]
__global__ __launch_bounds__(256) void k_scores(const float* __restrict__ dsL,
                                                const float* __restrict__ dsR,
                                                const float* __restrict__ dsM,
                                                const float* __restrict__ denL,
                                                const float* __restrict__ denR,
                                                float* __restrict__ scL,
                                                float* __restrict__ scR) {
    __shared__ __align__(16) float Ap[64][LDP];   // [l-in-tile][k] transposed panel
    __shared__ __align__(16) float Bp[64][LDP];   // [p-in-tile][k]
    const float* dsF = blockIdx.y ? dsR : dsL;
    const float* den = blockIdx.y ? denR : denL;
    float* out       = blockIdx.y ? scR : scL;
    int b   = blockIdx.z;
    int l0  = (blockIdx.x / 9) * 64;
    int p0  = (blockIdx.x % 9) * 64;
    int tid = threadIdx.x;
    int lane = tid & 31, w = tid >> 5;
    int tr  = w >> 1;            // 0..3 : 16-row l tile
    int tcb = (w & 1) * 2;       // 0/2  : pair of 16-col p tiles
    int m16   = lane & 15;
    int khalf = (lane >> 4) * 2; // lanes 16-31 hold K=2,3 per ISA A/B layout
    v8f acc0 = {}; v8f acc1 = {};

    for (int kb = 0; kb < K9; kb += TK) {
        __syncthreads();
        #pragma unroll
        for (int t = 0; t < 8; ++t) {
            int idx = tid + t * 256;   // 0..2047
            int lr  = idx & 63;        // fastest: contiguous patch index -> coalesced x
            int kk  = idx >> 6;        // 0..31
            Ap[lr][kk] = patch9(dsF, b, kb + kk, l0 + lr);
            Bp[lr][kk] = patch9(dsM, b, kb + kk, p0 + lr);
        }
        __syncthreads();
        #pragma unroll
        for (int kk0 = 0; kk0 < TK; kk0 += 4) {
            int ka = kk0 + khalf;
            v2f a  = *(const v2f*)&Ap[tr * 16 + m16][ka];
            v2f b0 = *(const v2f*)&Bp[tcb * 16 + m16][ka];
            v2f b1 = *(const v2f*)&Bp[(tcb + 1) * 16 + m16][ka];
            acc0 = __builtin_amdgcn_wmma_f32_16x16x4_f32(false, a, false, b0,
                                                         (short)0, acc0, false, false);
            acc1 = __builtin_amdgcn_wmma_f32_16x16x4_f32(false, a, false, b1,
                                                         (short)0, acc1, false, false);
        }
    }
    // epilogue: D layout — lane&15 = col, row m = v + 8*(lane>=16)
    int n     = lane & 15;
    int rbase = l0 + tr * 16 + ((lane >> 4) << 3);
    #pragma unroll
    for (int v = 0; v < 8; ++v) {
        int row = rbase + v;
        float scl = SCALE_ / den[b * L_ + row];
        out[((size_t)b * L_ + row) * L_ + p0 + tcb * 16 + n]       = acc0[v] * scl;
        out[((size_t)b * L_ + row) * L_ + p0 + (tcb + 1) * 16 + n] = acc1[v] * scl;
    }
}

// ---------------- kernel 4: softmax over the 576 filters (axis l) ----------------
__global__ __launch_bounds__(192) void k_softmax(float* __restrict__ scL,
                                                 float* __restrict__ scR) {
    float* sc = blockIdx.y ? scR : scL;
    int b = blockIdx.z;
    int p = blockIdx.x * 192 + threadIdx.x;          // 0..575 exactly
    float* col = sc + (size_t)b * L_ * L_ + p;
    float mx = -3.0e38f;
    for (int l = 0; l < L_; ++l) mx = fmaxf(mx, col[(size_t)l * L_]);
    float sum = 0.0f;
    for (int l = 0; l < L_; ++l) {
        float e = expf(col[(size_t)l * L_] - mx);
        col[(size_t)l * L_] = e;
        sum += e;
    }
    float inv = 1.0f / sum;
    for (int l = 0; l < L_; ++l) col[(size_t)l * L_] *= inv;
}

// ---------------- kernel 5: reconstruct GEMM (fp32 WMMA 16x16x4) ----------------
// Y[b, m=(c,i,j), pos] = sum_l raw[m,l] * attn[l,pos]
__global__ __launch_bounds__(256) void k_recon(const float* __restrict__ shcL,
                                               const float* __restrict__ shcR,
                                               const float* __restrict__ atL,
                                               const float* __restrict__ atR,
                                               float* __restrict__ YL,
                                               float* __restrict__ YR) {
    __shared__ __align__(16) float Ap[64][LDP];   // [m-in-tile][k(l)]
    __shared__ __align__(16) float Bp[64][LDP];   // [pos-in-tile][k(l)]
    const float* sc = blockIdx.y ? shcR : shcL;
    const float* at = blockIdx.y ? atR  : atL;
    float* Y        = blockIdx.y ? YR   : YL;
    int b   = blockIdx.z;
    int m0  = (blockIdx.x / 9) * 64;
    int p0  = (blockIdx.x % 9) * 64;
    int tid = threadIdx.x;
    int lane = tid & 31, w = tid >> 5;
    int tr  = w >> 1;
    int tcb = (w & 1) * 2;
    int m16   = lane & 15;
    int khalf = (lane >> 4) * 2;
    v8f acc0 = {}; v8f acc1 = {};

    for (int kb = 0; kb < L_; kb += TK) {
        __syncthreads();
        #pragma unroll
        for (int t = 0; t < 8; ++t) {
            int idx = tid + t * 256;
            int lr  = idx & 63;
            int kk  = idx >> 6;
            Ap[lr][kk] = patch16(sc, b, m0 + lr, kb + kk);
            Bp[lr][kk] = at[((size_t)b * L_ + kb + kk) * L_ + p0 + lr]; // coalesced over lr
        }
        __syncthreads();
        #pragma unroll
        for (int kk0 = 0; kk0 < TK; kk0 += 4) {
            int ka = kk0 + khalf;
            v2f a  = *(const v2f*)&Ap[tr * 16 + m16][ka];
            v2f b0 = *(const v2f*)&Bp[tcb * 16 + m16][ka];
            v2f b1 = *(const v2f*)&Bp[(tcb + 1) * 16 + m16][ka];
            acc0 = __builtin_amdgcn_wmma_f32_16x16x4_f32(false, a, false, b0,
                                                         (short)0, acc0, false, false);
            acc1 = __builtin_amdgcn_wmma_f32_16x16x4_f32(false, a, false, b1,
                                                         (short)0, acc1, false, false);
        }
    }
    int n     = lane & 15;
    int rbase = m0 + tr * 16 + ((lane >> 4) << 3);
    #pragma unroll
    for (int v = 0; v < 8; ++v) {
        int row = rbase + v;
        Y[((size_t)b * K16 + row) * L_ + p0 + tcb * 16 + n]       = acc0[v];
        Y[((size_t)b * K16 + row) * L_ + p0 + (tcb + 1) * 16 + n] = acc1[v];
    }
}

// ---------------- kernel 6: transpose-conv overlap gather + /4 + cosine blend ----
__global__ __launch_bounds__(256) void k_finalize(const float* __restrict__ YL,
                                                  const float* __restrict__ YR,
                                                  float* __restrict__ out) {
    int idx = blockIdx.x * 256 + threadIdx.x;       // B*C*48*48
    if (idx >= B_ * C_ * H_ * W_) return;
    int s = idx % W_; int t = idx / W_;
    int r = t % H_;  t /= H_;
    int c = t % C_;  int b = t / C_;
    float aL = 0.0f, aR = 0.0f;
    int i0 = (r + 1) & 1, j0 = (s + 1) & 1;
    #pragma unroll
    for (int di = 0; di < 2; ++di) {
        int i = i0 + 2 * di, ry = r + 1 - i;
        if (ry < 0 || ry >= 2 * HD) continue;
        int y = ry >> 1;
        #pragma unroll
        for (int dj = 0; dj < 2; ++dj) {
            int j = j0 + 2 * dj, sx = s + 1 - j;
            if (sx < 0 || sx >= 2 * WD) continue;
            int x = sx >> 1;
            int m = ((c << 2) | i) * 4 + j;
            size_t off = ((size_t)b * K16 + m) * L_ + y * WD + x;
            aL += YL[off];
            aR += YR[off];
        }
    }
    const float PI = 3.14159265358979323846f;
    float wl = 0.5f * (cosf(PI * (float)s / (float)(W_ - 1)) + 1.0f);
    float wr = 0.5f * (cosf(PI * (float)(W_ - 1 - s) / (float)(W_ - 1)) + 1.0f);
    out[idx] = 0.25f * (wl * aL + wr * aR);
}

// ---------------- launcher ----------------
extern "C" void kernel_launch(void* const* d_in, const int* in_sizes, int n_in,
                              void* d_out, int out_size, void* d_ws, size_t ws_size,
                              hipStream_t stream) {
    (void)in_sizes; (void)n_in; (void)out_size; (void)ws_size;
    const float* left  = (const float*)d_in[0];
    const float* right = (const float*)d_in[1];
    const float* mid   = (const float*)d_in[2];
    const float* shcL  = (const float*)d_in[3];
    const float* shcR  = (const float*)d_in[4];
    float* out = (float*)d_out;

    // workspace carve-up (floats); total ~46.6 MB
    float* w   = (float*)d_ws;
    const size_t NDS = (size_t)B_ * C_ * HD * WD;    // 294912
    const size_t NDN = (size_t)B_ * L_;              // 1152
    const size_t NSC = (size_t)B_ * L_ * L_;         // 663552
    const size_t NY  = (size_t)B_ * K16 * L_;        // 4718592
    float* dsL  = w;
    float* dsR  = dsL + NDS;
    float* dsM  = dsR + NDS;
    float* denL = dsM + NDS;
    float* denR = denL + NDN;
    float* scL  = denR + NDN;
    float* scR  = scL + NSC;
    float* YL   = scR + NSC;
    float* YR   = YL + NY;

    // 1) downsample left/right/mid by 2 (nearest)
    k_prep_ds<<<dim3((NDS + 255) / 256), dim3(256), 0, stream>>>(left, right, mid,
                                                                 dsL, dsR, dsM);
    // 2) feature-patch L2 denominators (one wave per patch)
    k_denom<<<dim3(B_ * L_ / 8, 2), dim3(256), 0, stream>>>(dsL, dsR, denL, denR);
    // 3) scores GEMM: [576 x 2304] x [2304 x 576], fp32 WMMA
    k_scores<<<dim3(81, 2, B_), dim3(256), 0, stream>>>(dsL, dsR, dsM,
                                                        denL, denR, scL, scR);
    // 4) softmax over filters
    k_softmax<<<dim3(3, 2, B_), dim3(192), 0, stream>>>(scL, scR);
    // 5) reconstruct GEMM: [4096 x 576] x [576 x 576], fp32 WMMA
    k_recon<<<dim3(576, 2, B_), dim3(256), 0, stream>>>(shcL, shcR, scL, scR, YL, YR);
    // 6) overlap gather, /4, cosine blend
    k_finalize<<<dim3((B_ * C_ * H_ * W_ + 255) / 256), dim3(256), 0, stream>>>(YL, YR, out);
}